// GNN_30374008718130
// MI455X (gfx1250) — compile-verified
//
#include <hip/hip_runtime.h>
#include <math.h>

#define EPSB   1e-5f
#define NNODES 64000
#define NEDGES 512000
#define NBATCH 64

typedef __attribute__((ext_vector_type(2))) float v2f;
typedef __attribute__((ext_vector_type(8))) float v8f;

// ---------------------------------------------------------------------------
// f32 WMMA 16x16x4 operand layout (ISA 7.12.2):
//   A: lane 0-15 -> M=lane, VGPRs hold K0/K1 ; lanes 16-31 -> K2/K3
//   B mirrors A with N (weight row) in place of M.
//   C/D: VGPR i -> (M = i + 8*half, N = lane&15)
// All leading dims are even -> per-lane float2 loads are 8B aligned.
// Both helpers are software double-buffered: loads for k-step k+4 are issued
// before the WMMAs of step k, so each WMMA only waits on iteration-old loads.
// ---------------------------------------------------------------------------

// One wave: 16 rows x (R*16) cols. Per k-step: 1 A load + R B loads, R WMMAs.
template <int R>
__device__ __forceinline__ void wmma_rowblock(const float* __restrict__ A, int lda,
                                              const float* __restrict__ B, int ldb,
                                              int K, v8f (&acc)[R]) {
  const int lane = threadIdx.x & 31;
  const int hf   = lane >> 4;
  const int lm   = lane & 15;
  const float* ap = A + (long)lm * lda + 2 * hf;
  const float* bp = B + (long)lm * ldb + 2 * hf;

  v2f a_cur = *(const v2f*)(ap);
  v2f b_cur[R];
#pragma unroll
  for (int r = 0; r < R; ++r) b_cur[r] = *(const v2f*)(bp + (long)(r * 16) * ldb);

  for (int k = 4; k < K; k += 4) {
    v2f a_nxt = *(const v2f*)(ap + k);
    v2f b_nxt[R];
#pragma unroll
    for (int r = 0; r < R; ++r) b_nxt[r] = *(const v2f*)(bp + (long)(r * 16) * ldb + k);
#pragma unroll
    for (int r = 0; r < R; ++r)
      acc[r] = __builtin_amdgcn_wmma_f32_16x16x4_f32(false, a_cur, false, b_cur[r],
                                                     (short)0, acc[r], false, false);
    a_cur = a_nxt;
#pragma unroll
    for (int r = 0; r < R; ++r) b_cur[r] = b_nxt[r];
  }
#pragma unroll
  for (int r = 0; r < R; ++r)
    acc[r] = __builtin_amdgcn_wmma_f32_16x16x4_f32(false, a_cur, false, b_cur[r],
                                                   (short)0, acc[r], false, false);
}

// One wave: (MB*16) rows x 16 cols. Per k-step: 1 B load + MB A loads, MB WMMAs.
template <int MB>
__device__ __forceinline__ void wmma_colblock(const float* __restrict__ A, int lda,
                                              const float* __restrict__ B, int ldb,
                                              int K, v8f (&acc)[MB]) {
  const int lane = threadIdx.x & 31;
  const int hf   = lane >> 4;
  const int lm   = lane & 15;
  const float* ap = A + (long)lm * lda + 2 * hf;
  const float* bp = B + (long)lm * ldb + 2 * hf;

  v2f b_cur = *(const v2f*)(bp);
  v2f a_cur[MB];
#pragma unroll
  for (int m = 0; m < MB; ++m) a_cur[m] = *(const v2f*)(ap + (long)(m * 16) * lda);

  for (int k = 4; k < K; k += 4) {
    v2f b_nxt = *(const v2f*)(bp + k);
    v2f a_nxt[MB];
#pragma unroll
    for (int m = 0; m < MB; ++m) a_nxt[m] = *(const v2f*)(ap + (long)(m * 16) * lda + k);
#pragma unroll
    for (int m = 0; m < MB; ++m)
      acc[m] = __builtin_amdgcn_wmma_f32_16x16x4_f32(false, a_cur[m], false, b_cur,
                                                     (short)0, acc[m], false, false);
    b_cur = b_nxt;
#pragma unroll
    for (int m = 0; m < MB; ++m) a_cur[m] = a_nxt[m];
  }
#pragma unroll
  for (int m = 0; m < MB; ++m)
    acc[m] = __builtin_amdgcn_wmma_f32_16x16x4_f32(false, a_cur[m], false, b_cur,
                                                   (short)0, acc[m], false, false);
}

// ---------------------------------------------------------------------------
__global__ void zero_f32(float* __restrict__ p, long n) {
  long i = (long)blockIdx.x * blockDim.x + threadIdx.x;
  if (i < n) p[i] = 0.f;
}

// Edge message + scatter-add: agg[dst] += h[src] * w (float4 per thread).
// h/agg fit in the 192MB L2 -> gathers + atomics stay on-chip.
__global__ void edge_scatter(const float* __restrict__ h, const int* __restrict__ src,
                             const int* __restrict__ dst, const float* __restrict__ w,
                             float* __restrict__ agg, int d) {
  const int nvec = d >> 2;
  long t = (long)blockIdx.x * blockDim.x + threadIdx.x;
  if (t >= (long)NEDGES * nvec) return;
  int e = (int)(t / nvec);
  int c = (int)(t - (long)e * nvec) << 2;
  int s  = src[e];
  int dd = dst[e];
  float we = w[e];
  const float4 hv = *(const float4*)(h + (long)s * d + c);
  float* ap = agg + (long)dd * d + c;
  atomicAdd(ap + 0, hv.x * we);
  atomicAdd(ap + 1, hv.y * we);
  atomicAdd(ap + 2, hv.z * we);
  atomicAdd(ap + 3, hv.w * we);
}

// out = relu(agg @ Wrel^T + h @ Wroot^T + b).
// One wave per 16-row M tile; R = dout/16 accumulators cover all columns.
// Grid exact (4000 waves) -> EXEC all ones for every WMMA.
template <int R>
__global__ void conv_gemm_relu(const float* __restrict__ agg, const float* __restrict__ h,
                               const float* __restrict__ Wrel, const float* __restrict__ Wroot,
                               const float* __restrict__ bias, float* __restrict__ out,
                               int din) {
  const int dout = R * 16;
  const int wid = blockIdx.x * (blockDim.x >> 5) + (threadIdx.x >> 5);
  const int m0 = wid << 4;

  v8f z = {};
  v8f acc[R];
#pragma unroll
  for (int r = 0; r < R; ++r) acc[r] = z;

  wmma_rowblock<R>(agg + (long)m0 * din, din, Wrel, din, din, acc);
  wmma_rowblock<R>(h   + (long)m0 * din, din, Wroot, din, din, acc);

  const int lane = threadIdx.x & 31, hf = lane >> 4, lm = lane & 15;
#pragma unroll
  for (int r = 0; r < R; ++r) {
    const float b = bias[r * 16 + lm];
#pragma unroll
    for (int i = 0; i < 8; ++i) {
      float v = acc[r][i] + b;
      out[(long)(m0 + i + 8 * hf) * dout + r * 16 + lm] = v > 0.f ? v : 0.f;
    }
  }
}

// BatchNorm (training) stats over N=64000 rows: one block per feature.
__global__ void bn_stats(const float* __restrict__ h, int d,
                         const float* __restrict__ gamma,
                         float* __restrict__ mean, float* __restrict__ scale) {
  const int f = blockIdx.x;
  float s = 0.f, s2 = 0.f;
  for (int i = threadIdx.x; i < NNODES; i += blockDim.x) {
    float v = h[(long)i * d + f];
    s += v; s2 += v * v;
  }
  __shared__ float sh[256], sh2[256];
  sh[threadIdx.x] = s; sh2[threadIdx.x] = s2;
  __syncthreads();
  for (int off = 128; off > 0; off >>= 1) {
    if ((int)threadIdx.x < off) {
      sh[threadIdx.x]  += sh[threadIdx.x + off];
      sh2[threadIdx.x] += sh2[threadIdx.x + off];
    }
    __syncthreads();
  }
  if (threadIdx.x == 0) {
    float mu  = sh[0] * (1.f / NNODES);
    float var = sh2[0] * (1.f / NNODES) - mu * mu;
    mean[f]  = mu;
    scale[f] = gamma[f] * rsqrtf(var + EPSB);
  }
}

__global__ void bn_apply(float* __restrict__ h, const float* __restrict__ mean,
                         const float* __restrict__ scale, const float* __restrict__ beta,
                         int dmask, long n) {
  long i = (long)blockIdx.x * blockDim.x + threadIdx.x;
  if (i >= n) return;
  int f = (int)i & dmask;   // d is a power of two (64 or 128)
  h[i] = (h[i] - mean[f]) * scale[f] + beta[f];
}

// FC1: [64,64000] x W[1024,64000]^T. K split across grid.y, f32 atomic reduce.
// One wave covers all 4 M tiles for its 16-column slice (B shared 4x).
__global__ void fc_gemm_atomic(const float* __restrict__ A, const float* __restrict__ W,
                               float* __restrict__ outRaw, int Nout, int K, int kChunk) {
  const int n0 = (blockIdx.x * (blockDim.x >> 5) + (threadIdx.x >> 5)) << 4;
  const int k0 = blockIdx.y * kChunk;
  v8f z = {};
  v8f acc[4];
#pragma unroll
  for (int m = 0; m < 4; ++m) acc[m] = z;
  wmma_colblock<4>(A + k0, K, W + (long)n0 * K + k0, K, kChunk, acc);
  const int lane = threadIdx.x & 31, hf = lane >> 4, lm = lane & 15;
#pragma unroll
  for (int m = 0; m < 4; ++m)
#pragma unroll
    for (int i = 0; i < 8; ++i)
      atomicAdd(&outRaw[(m * 16 + i + 8 * hf) * Nout + n0 + lm], acc[m][i]);
}

// Small GEMM, direct store of raw (pre-bias) result. One wave = 64 x 16 slab.
__global__ void fc_gemm_plain(const float* __restrict__ A, const float* __restrict__ W,
                              float* __restrict__ outRaw, int Nout, int K) {
  const int n0 = (blockIdx.x * (blockDim.x >> 5) + (threadIdx.x >> 5)) << 4;
  v8f z = {};
  v8f acc[4];
#pragma unroll
  for (int m = 0; m < 4; ++m) acc[m] = z;
  wmma_colblock<4>(A, K, W + (long)n0 * K, K, K, acc);
  const int lane = threadIdx.x & 31, hf = lane >> 4, lm = lane & 15;
#pragma unroll
  for (int m = 0; m < 4; ++m)
#pragma unroll
    for (int i = 0; i < 8; ++i)
      outRaw[(m * 16 + i + 8 * hf) * Nout + n0 + lm] = acc[m][i];
}

// Fused bias + ReLU + BatchNorm over the 64-row batch dim: thread per feature.
__global__ void fc_post(const float* __restrict__ yraw, const float* __restrict__ bias,
                        const float* __restrict__ gamma, const float* __restrict__ beta,
                        float* __restrict__ y, int d) {
  int f = blockIdx.x * blockDim.x + threadIdx.x;
  if (f >= d) return;
  float b = bias[f], s = 0.f, s2 = 0.f;
  for (int r = 0; r < NBATCH; ++r) {
    float v = yraw[r * d + f] + b;
    v = v > 0.f ? v : 0.f;
    s += v; s2 += v * v;
  }
  float mu  = s * (1.f / NBATCH);
  float var = s2 * (1.f / NBATCH) - mu * mu;
  float sc  = gamma[f] * rsqrtf(var + EPSB);
  float be  = beta[f];
  for (int r = 0; r < NBATCH; ++r) {
    float v = yraw[r * d + f] + b;
    v = v > 0.f ? v : 0.f;
    y[r * d + f] = (v - mu) * sc + be;
  }
}

// Output layer: GEMM + bias + sigmoid. One wave (block of 32) per 64x16 slab.
__global__ void out_gemm_sigmoid(const float* __restrict__ A, const float* __restrict__ W,
                                 const float* __restrict__ bias, float* __restrict__ out,
                                 int Nout, int K) {
  const int n0 = blockIdx.x << 4;
  v8f z = {};
  v8f acc[4];
#pragma unroll
  for (int m = 0; m < 4; ++m) acc[m] = z;
  wmma_colblock<4>(A, K, W + (long)n0 * K, K, K, acc);
  const int lane = threadIdx.x & 31, hf = lane >> 4, lm = lane & 15;
  const float b = bias[n0 + lm];
#pragma unroll
  for (int m = 0; m < 4; ++m)
#pragma unroll
    for (int i = 0; i < 8; ++i) {
      float v = acc[m][i] + b;
      out[(m * 16 + i + 8 * hf) * Nout + n0 + lm] = 1.f / (1.f + expf(-v));
    }
}

// ---------------------------------------------------------------------------
extern "C" void kernel_launch(void* const* d_in, const int* in_sizes, int n_in,
                              void* d_out, int out_size, void* d_ws, size_t ws_size,
                              hipStream_t stream) {
  (void)in_sizes; (void)n_in; (void)out_size; (void)ws_size;

  const float* x   = (const float*)d_in[0];
  const int*   ei  = (const int*)d_in[1];
  const float* ea  = (const float*)d_in[2];
  const float* Wrel0  = (const float*)d_in[3];
  const float* brel0  = (const float*)d_in[4];
  const float* Wroot0 = (const float*)d_in[5];
  const float* Wrel1  = (const float*)d_in[6];
  const float* brel1  = (const float*)d_in[7];
  const float* Wroot1 = (const float*)d_in[8];
  const float* Wrel2  = (const float*)d_in[9];
  const float* brel2  = (const float*)d_in[10];
  const float* Wroot2 = (const float*)d_in[11];
  const float* g0 = (const float*)d_in[12]; const float* be0 = (const float*)d_in[13];
  const float* g1 = (const float*)d_in[14]; const float* be1 = (const float*)d_in[15];
  const float* g2 = (const float*)d_in[16]; const float* be2 = (const float*)d_in[17];
  const float* Wfc0 = (const float*)d_in[18]; const float* bfc0 = (const float*)d_in[19];
  const float* gfc0 = (const float*)d_in[20]; const float* befc0 = (const float*)d_in[21];
  const float* Wfc1 = (const float*)d_in[22]; const float* bfc1 = (const float*)d_in[23];
  const float* gfc1 = (const float*)d_in[24]; const float* befc1 = (const float*)d_in[25];
  const float* outW = (const float*)d_in[26]; const float* outb = (const float*)d_in[27];
  float* out = (float*)d_out;

  const int* src = ei;
  const int* dst = ei + NEDGES;

  // Workspace layout (~99 MB of fp32).
  float* hA    = (float*)d_ws;                    // [64000,128] max
  float* hB    = hA + (long)NNODES * 128;
  float* agg   = hB + (long)NNODES * 128;
  float* y1r   = agg + (long)NNODES * 128;        // [64,1024]
  float* y1    = y1r + 64 * 1024;
  float* y2r   = y1 + 64 * 1024;                  // [64,512]
  float* y2    = y2r + 64 * 512;
  float* mean  = y2 + 64 * 512;                   // [<=1024]
  float* scale = mean + 1024;

  // ---- Conv layer 0: din=4 -> dout=64 (input is x) ----
  zero_f32<<<((long)NNODES * 4 + 255) / 256, 256, 0, stream>>>(agg, (long)NNODES * 4);
  edge_scatter<<<(NEDGES + 255) / 256, 256, 0, stream>>>(x, src, dst, ea, agg, 4);
  conv_gemm_relu<4><<<500, 256, 0, stream>>>(agg, x, Wrel0, Wroot0, brel0, hA, 4);
  bn_stats<<<64, 256, 0, stream>>>(hA, 64, g0, mean, scale);
  bn_apply<<<((long)NNODES * 64 + 255) / 256, 256, 0, stream>>>(hA, mean, scale, be0, 63,
                                                                (long)NNODES * 64);

  // ---- Conv layer 1: 64 -> 128 ----
  zero_f32<<<((long)NNODES * 64 + 255) / 256, 256, 0, stream>>>(agg, (long)NNODES * 64);
  edge_scatter<<<((long)NEDGES * 16 + 255) / 256, 256, 0, stream>>>(hA, src, dst, ea, agg, 64);
  conv_gemm_relu<8><<<500, 256, 0, stream>>>(agg, hA, Wrel1, Wroot1, brel1, hB, 64);
  bn_stats<<<128, 256, 0, stream>>>(hB, 128, g1, mean, scale);
  bn_apply<<<((long)NNODES * 128 + 255) / 256, 256, 0, stream>>>(hB, mean, scale, be1, 127,
                                                                 (long)NNODES * 128);

  // ---- Conv layer 2: 128 -> 64 ----
  zero_f32<<<((long)NNODES * 128 + 255) / 256, 256, 0, stream>>>(agg, (long)NNODES * 128);
  edge_scatter<<<((long)NEDGES * 32 + 255) / 256, 256, 0, stream>>>(hB, src, dst, ea, agg, 128);
  conv_gemm_relu<4><<<500, 256, 0, stream>>>(agg, hB, Wrel2, Wroot2, brel2, hA, 128);
  bn_stats<<<64, 256, 0, stream>>>(hA, 64, g2, mean, scale);
  bn_apply<<<((long)NNODES * 64 + 255) / 256, 256, 0, stream>>>(hA, mean, scale, be2, 63,
                                                                (long)NNODES * 64);

  // ---- FC1: reshape is a reinterpret; A = hA as [64, 64000] ----
  zero_f32<<<(64 * 1024 + 255) / 256, 256, 0, stream>>>(y1r, 64 * 1024);
  dim3 gfc(1024 / (16 * 8), 16);  // 8 N-tiles per block x 16 K parts (chunk = 4000)
  fc_gemm_atomic<<<gfc, 256, 0, stream>>>(hA, Wfc0, y1r, 1024, 64000, 4000);
  fc_post<<<(1024 + 255) / 256, 256, 0, stream>>>(y1r, bfc0, gfc0, befc0, y1, 1024);

  // ---- FC2: [64,1024] x [512,1024]^T ----
  fc_gemm_plain<<<512 / (16 * 8), 256, 0, stream>>>(y1, Wfc1, y2r, 512, 1024);
  fc_post<<<(512 + 255) / 256, 256, 0, stream>>>(y2r, bfc1, gfc1, befc1, y2, 512);

  // ---- Output: [64,512] x [2000,512]^T + sigmoid ----
  out_gemm_sigmoid<<<125, 32, 0, stream>>>(y2, outW, outb, out, 2000, 512);
}